// LlamaAttention_81037442941430
// MI455X (gfx1250) — compile-verified
//
#include <hip/hip_runtime.h>

// ---------------------------------------------------------------- constants
static constexpr int B  = 2;
static constexpr int S  = 2048;
static constexpr int D  = 2048;
static constexpr int H  = 32;
static constexpr int KV = 8;
static constexpr int HD = 64;
static constexpr float NEG_INF = -1.0e9f;

static constexpr int LDSPITCH = 20;   // GEMM tile row pitch (80B: 16B-aligned, bank-safe)
static constexpr int KVPITCH  = 68;   // attn K/V tile row pitch (272B: 16B-aligned, bank-safe)

typedef __attribute__((ext_vector_type(2))) float v2f;
typedef __attribute__((ext_vector_type(8))) float v8f;

// D = A(16x4, f32) * B(4x16, f32) + C(16x16, f32)   -- native fp32 WMMA
static __device__ __forceinline__ v8f wmma_f32(v2f a, v2f b, v8f c) {
    return __builtin_amdgcn_wmma_f32_16x16x4_f32(
        /*neg_a=*/false, a, /*neg_b=*/false, b,
        /*c_mod=*/(short)0, c, /*reuse_a=*/false, /*reuse_b=*/false);
}

// Async memory -> LDS, 16 bytes per lane (GLOBAL_LOAD_ASYNC_TO_LDS_B128, ASYNCcnt).
static __device__ __forceinline__ void async_copy16(unsigned lds_off, const float* gp) {
    asm volatile("global_load_async_to_lds_b128 %0, %1, off"
                 :: "v"(lds_off), "v"((unsigned long long)(size_t)gp)
                 : "memory");
}
template <int N>
static __device__ __forceinline__ void wait_async() {
    asm volatile("s_wait_asynccnt %0" :: "i"(N) : "memory");
}

// =======================================================================
// Kernel 1: q = x @ wq^T, RoPE fused, writes q_rope [B][H][S][HD].
// Block = 8 waves -> 64(M) x 256(N); wave -> 32 x 64.
// K staged in 16-wide chunks, double-buffered async-to-LDS.
// =======================================================================
__global__ __launch_bounds__(256)
void qproj_rope_kernel(const float* __restrict__ x,
                       const float* __restrict__ wq,
                       const float* __restrict__ cosb,
                       const float* __restrict__ sinb,
                       float* __restrict__ q_rope)
{
    __shared__ float aLds[2][64 * LDSPITCH];
    __shared__ float bLds[2][256 * LDSPITCH];

    const int t    = threadIdx.x;
    const int lane = t & 31;
    const int wib  = t >> 5;
    const int half = lane >> 4;
    const int l16  = lane & 15;
    const int mt   = blockIdx.x >> 3;          // 64 M-blocks
    const int ntb  = blockIdx.x & 7;           // 8 N-blocks
    const int row0 = mt * 64;
    const int col0 = ntb * 256;
    const int wm   = wib >> 2;                 // 0..1
    const int wn   = wib & 3;                  // 0..3

    const int lr = t >> 2;                     // 0..63
    const int lc = (t & 3) << 2;               // 0,4,8,12

    v8f acc[2][4];
    #pragma unroll
    for (int i = 0; i < 2; ++i)
        #pragma unroll
        for (int p = 0; p < 4; ++p) acc[i][p] = v8f{};

    unsigned aoff[2], boff[2][4];
    #pragma unroll
    for (int s2 = 0; s2 < 2; ++s2) {
        aoff[s2] = (unsigned)(size_t)&aLds[s2][lr * LDSPITCH + lc];
        #pragma unroll
        for (int p = 0; p < 4; ++p)
            boff[s2][p] = (unsigned)(size_t)&bLds[s2][(lr + 64 * p) * LDSPITCH + lc];
    }

    const float* agp = x + (size_t)(row0 + lr) * D + lc;
    const float* bgp[4];
    #pragma unroll
    for (int p = 0; p < 4; ++p)
        bgp[p] = wq + (size_t)(col0 + lr + 64 * p) * D + lc;

    constexpr int NC = D / 16;                 // 128 chunks
    // prologue: stage chunk 0 into buffer 0
    async_copy16(aoff[0], agp);
    #pragma unroll
    for (int p = 0; p < 4; ++p) async_copy16(boff[0][p], bgp[p]);

    for (int c = 0; c < NC; ++c) {
        const int cur = c & 1;
        if (c + 1 < NC) {                      // stage next chunk into other buffer
            async_copy16(aoff[cur ^ 1], agp + (size_t)(c + 1) * 16);
            #pragma unroll
            for (int p = 0; p < 4; ++p)
                async_copy16(boff[cur ^ 1][p], bgp[p] + (size_t)(c + 1) * 16);
            wait_async<5>();                   // only newest 5 may remain
        } else {
            wait_async<0>();
        }
        __syncthreads();

        const float* aL = &aLds[cur][0];
        const float* bL = &bLds[cur][0];
        #pragma unroll
        for (int ks = 0; ks < 4; ++ks) {
            const int ko = ks * 4 + 2 * half;
            v2f af[2];
            #pragma unroll
            for (int i = 0; i < 2; ++i) {
                const int r = (wm * 32 + i * 16 + l16) * LDSPITCH + ko;
                af[i][0] = aL[r]; af[i][1] = aL[r + 1];
            }
            v2f bf[4];
            #pragma unroll
            for (int p = 0; p < 4; ++p) {
                const int r = (wn * 64 + p * 16 + l16) * LDSPITCH + ko;
                bf[p][0] = bL[r]; bf[p][1] = bL[r + 1];
            }
            #pragma unroll
            for (int i = 0; i < 2; ++i)
                #pragma unroll
                for (int p = 0; p < 4; ++p)
                    acc[i][p] = wmma_f32(af[i], bf[p], acc[i][p]);
        }
        __syncthreads();                       // done reading before it is re-staged
    }

    // RoPE epilogue: column pairs (2i,2i+1) are adjacent lanes in C/D layout.
    const int h = ntb * 4 + wn;                // 64-col span == one head
    #pragma unroll
    for (int i = 0; i < 2; ++i) {
        #pragma unroll
        for (int p = 0; p < 4; ++p) {
            const int d  = p * 16 + l16;       // 0..63 within head
            const int ii = d >> 1;
            #pragma unroll
            for (int v = 0; v < 8; ++v) {
                const int grow = row0 + wm * 32 + i * 16 + v + 8 * half;
                const int bi   = grow >> 11;          // / S
                const int s    = grow & (S - 1);
                const float val   = acc[i][p][v];
                const float other = __shfl_xor(val, 1, 32);
                const float c  = cosb[s * (HD / 2) + ii];
                const float sn = sinb[s * (HD / 2) + ii];
                const float r  = ((d & 1) == 0) ? (val * c - other * sn)
                                                : (other * sn + val * c);
                q_rope[(((size_t)bi * H + h) * S + s) * HD + d] = r;
            }
        }
    }
}

// =======================================================================
// Kernel 2: RoPE on k_cache [B][S][KV][HD] -> k_rope [B][KV][S][HD]
// =======================================================================
__global__ __launch_bounds__(256)
void krope_kernel(const float* __restrict__ kc,
                  const float* __restrict__ cosb,
                  const float* __restrict__ sinb,
                  float* __restrict__ k_rope)
{
    const int tid = blockIdx.x * 256 + threadIdx.x;   // B*S*KV*(HD/2) = 2^20
    const int i  = tid & 31;
    const int kv = (tid >> 5) & 7;
    const int s  = (tid >> 8) & (S - 1);
    const int b  = tid >> 19;

    const float* src = kc + (((size_t)(b * S + s) * KV + kv) * HD);
    const float a  = src[2 * i];
    const float bb = src[2 * i + 1];
    const float c  = cosb[s * (HD / 2) + i];
    const float sn = sinb[s * (HD / 2) + i];
    float* dst = k_rope + ((((size_t)b * KV + kv) * S + s) * HD);
    dst[2 * i]     = a * c - bb * sn;
    dst[2 * i + 1] = a * sn + bb * c;
}

// =======================================================================
// Kernel 3: flash attention. Block = 8 waves sharing (b, h, kv);
// wave w handles q-tile qb+w. K/V tiles staged once per block per j,
// double-buffered async-to-LDS; Q fragments hoisted to registers.
// =======================================================================
__global__ __launch_bounds__(256)
void attn_kernel(const float* __restrict__ q_rope,
                 const float* __restrict__ k_rope,
                 const float* __restrict__ v_cache,
                 float* __restrict__ attn_out)
{
    __shared__ float kLds[2][16 * KVPITCH];
    __shared__ float vLds[2][16 * KVPITCH];
    __shared__ float pbuf[8][16 * 17];

    const int t    = threadIdx.x;
    const int lane = t & 31;
    const int wib  = t >> 5;
    const int half = lane >> 4;
    const int l16  = lane & 15;

    const int wave0 = blockIdx.x * 8;          // 8 | 128 => no &127 wrap in-block
    const int qb   = wave0 & 127;              // block-uniform base q-tile
    const int h    = (wave0 >> 7) & 31;        // block-uniform head
    const int b    = wave0 >> 12;              // block-uniform batch
    const int kv   = h >> 2;                   // GQA: 4 heads share a kv head
    const int qt   = qb + wib;                 // this wave's q-tile
    const int qs   = qt << 4;
    const int jmax = qb + 7;                   // block-uniform loop bound

    const float* qbase = q_rope + ((size_t)b * H + h) * S * HD;
    const float* kbase = k_rope + ((size_t)b * KV + kv) * S * HD;
    const float* vbase = v_cache + ((size_t)b * S * KV + kv) * HD;

    // ---- staging lanes: 16B per thread covers a 16x64 tile ----
    const int lr16 = t >> 4;                   // 0..15 (key row in tile)
    const int lc16 = (t & 15) << 2;            // 0..60 (d column)
    unsigned koff[2], voff[2];
    #pragma unroll
    for (int s2 = 0; s2 < 2; ++s2) {
        koff[s2] = (unsigned)(size_t)&kLds[s2][lr16 * KVPITCH + lc16];
        voff[s2] = (unsigned)(size_t)&vLds[s2][lr16 * KVPITCH + lc16];
    }

    // ---- hoist this wave's Q fragments (16 rows x 64 d) into registers ----
    v2f qf[16];
    {
        const float* qrow = qbase + (size_t)(qs + l16) * HD + 2 * half;
        #pragma unroll
        for (int ks = 0; ks < 16; ++ks) {
            qf[ks][0] = qrow[ks * 4];
            qf[ks][1] = qrow[ks * 4 + 1];
        }
    }

    float rowm[8], rowl[8];
    v8f o[4];
    #pragma unroll
    for (int v = 0; v < 8; ++v) { rowm[v] = -3.0e38f; rowl[v] = 0.0f; }
    #pragma unroll
    for (int tt = 0; tt < 4; ++tt) o[tt] = v8f{};

    const float scale = 0.125f;                // 1/sqrt(64)
    float* pl = &pbuf[wib][0];

    // prologue: stage j = 0
    async_copy16(koff[0], kbase + (size_t)lr16 * HD + lc16);
    async_copy16(voff[0], vbase + (size_t)lr16 * KV * HD + lc16);

    for (int j = 0; j <= jmax; ++j) {
        const int cur = j & 1;
        if (j + 1 <= jmax) {
            async_copy16(koff[cur ^ 1],
                         kbase + (size_t)((j + 1) * 16 + lr16) * HD + lc16);
            async_copy16(voff[cur ^ 1],
                         vbase + (size_t)((j + 1) * 16 + lr16) * KV * HD + lc16);
            wait_async<2>();                   // only newest 2 may remain
        } else {
            wait_async<0>();
        }
        __syncthreads();

        if (j <= qt) {                         // wave-uniform guard; barriers outside
            const float* kl = &kLds[cur][0];
            const float* vl = &vLds[cur][0];

            // ---- S_tile = q (16xHD) @ k^T (HDx16) ----
            v8f sacc = v8f{};
            #pragma unroll
            for (int ks = 0; ks < 16; ++ks) {
                const int ko = ks * 4 + 2 * half;
                v2f bb; bb[0] = kl[l16 * KVPITCH + ko];
                        bb[1] = kl[l16 * KVPITCH + ko + 1];
                sacc = wmma_f32(qf[ks], bb, sacc);
            }

            // ---- scale + causal mask + online softmax ----
            float pv[8], corr[8];
            #pragma unroll
            for (int v = 0; v < 8; ++v) {
                float sc = sacc[v] * scale;
                const int qrowi = qs + v + 8 * half;
                const int key   = j * 16 + l16;
                sc = (key > qrowi) ? NEG_INF : sc;
                float mx = sc;                 // row-max over 16-lane half
                mx = fmaxf(mx, __shfl_xor(mx, 1, 32));
                mx = fmaxf(mx, __shfl_xor(mx, 2, 32));
                mx = fmaxf(mx, __shfl_xor(mx, 4, 32));
                mx = fmaxf(mx, __shfl_xor(mx, 8, 32));
                const float mnew = fmaxf(rowm[v], mx);
                const float c = __expf(rowm[v] - mnew);
                const float p = __expf(sc - mnew);
                float rs = p;                  // row-sum over 16-lane half
                rs += __shfl_xor(rs, 1, 32);
                rs += __shfl_xor(rs, 2, 32);
                rs += __shfl_xor(rs, 4, 32);
                rs += __shfl_xor(rs, 8, 32);
                rowl[v] = rowl[v] * c + rs;
                rowm[v] = mnew;
                corr[v] = c;
                pv[v]   = p;
            }
            #pragma unroll
            for (int tt = 0; tt < 4; ++tt)
                #pragma unroll
                for (int v = 0; v < 8; ++v) o[tt][v] *= corr[v];

            // ---- P (C/D layout) -> LDS -> A-layout fragments ----
            #pragma unroll
            for (int v = 0; v < 8; ++v)
                pl[(v + 8 * half) * 17 + l16] = pv[v];
            asm volatile("s_wait_dscnt 0x0" ::: "memory");

            // ---- O += P (16x16) @ V (16xHD) ----
            #pragma unroll
            for (int kb2 = 0; kb2 < 16; kb2 += 4) {
                v2f pa;
                pa[0] = pl[l16 * 17 + kb2 + 2 * half];
                pa[1] = pl[l16 * 17 + kb2 + 2 * half + 1];
                const int r0 = (kb2 + 2 * half) * KVPITCH;
                #pragma unroll
                for (int tt = 0; tt < 4; ++tt) {
                    v2f vb;
                    vb[0] = vl[r0 + tt * 16 + l16];
                    vb[1] = vl[r0 + KVPITCH + tt * 16 + l16];
                    o[tt] = wmma_f32(pa, vb, o[tt]);
                }
            }
            asm volatile("s_wait_dscnt 0x0" ::: "memory");
        }
        __syncthreads();                       // K/V buffer reuse fence
    }

    // ---- normalize and store: attn_out [B][S][H*HD] ----
    #pragma unroll
    for (int tt = 0; tt < 4; ++tt) {
        #pragma unroll
        for (int v = 0; v < 8; ++v) {
            const int s = qs + v + 8 * half;
            const float val = o[tt][v] / rowl[v];
            attn_out[((size_t)b * S + s) * D + h * 64 + tt * 16 + l16] = val;
        }
    }
}

// =======================================================================
// Kernel 4: out = attn_out @ wo^T  (double-buffered async LDS GEMM)
// =======================================================================
__global__ __launch_bounds__(256)
void oproj_kernel(const float* __restrict__ attn,
                  const float* __restrict__ wo,
                  float* __restrict__ out)
{
    __shared__ float aLds[2][64 * LDSPITCH];
    __shared__ float bLds[2][256 * LDSPITCH];

    const int t    = threadIdx.x;
    const int lane = t & 31;
    const int wib  = t >> 5;
    const int half = lane >> 4;
    const int l16  = lane & 15;
    const int mt   = blockIdx.x >> 3;
    const int ntb  = blockIdx.x & 7;
    const int row0 = mt * 64;
    const int col0 = ntb * 256;
    const int wm   = wib >> 2;
    const int wn   = wib & 3;

    const int lr = t >> 2;
    const int lc = (t & 3) << 2;

    v8f acc[2][4];
    #pragma unroll
    for (int i = 0; i < 2; ++i)
        #pragma unroll
        for (int p = 0; p < 4; ++p) acc[i][p] = v8f{};

    unsigned aoff[2], boff[2][4];
    #pragma unroll
    for (int s2 = 0; s2 < 2; ++s2) {
        aoff[s2] = (unsigned)(size_t)&aLds[s2][lr * LDSPITCH + lc];
        #pragma unroll
        for (int p = 0; p < 4; ++p)
            boff[s2][p] = (unsigned)(size_t)&bLds[s2][(lr + 64 * p) * LDSPITCH + lc];
    }

    const float* agp = attn + (size_t)(row0 + lr) * D + lc;
    const float* bgp[4];
    #pragma unroll
    for (int p = 0; p < 4; ++p)
        bgp[p] = wo + (size_t)(col0 + lr + 64 * p) * D + lc;

    constexpr int NC = D / 16;
    async_copy16(aoff[0], agp);
    #pragma unroll
    for (int p = 0; p < 4; ++p) async_copy16(boff[0][p], bgp[p]);

    for (int c = 0; c < NC; ++c) {
        const int cur = c & 1;
        if (c + 1 < NC) {
            async_copy16(aoff[cur ^ 1], agp + (size_t)(c + 1) * 16);
            #pragma unroll
            for (int p = 0; p < 4; ++p)
                async_copy16(boff[cur ^ 1][p], bgp[p] + (size_t)(c + 1) * 16);
            wait_async<5>();
        } else {
            wait_async<0>();
        }
        __syncthreads();

        const float* aL = &aLds[cur][0];
        const float* bL = &bLds[cur][0];
        #pragma unroll
        for (int ks = 0; ks < 4; ++ks) {
            const int ko = ks * 4 + 2 * half;
            v2f af[2];
            #pragma unroll
            for (int i = 0; i < 2; ++i) {
                const int r = (wm * 32 + i * 16 + l16) * LDSPITCH + ko;
                af[i][0] = aL[r]; af[i][1] = aL[r + 1];
            }
            v2f bf[4];
            #pragma unroll
            for (int p = 0; p < 4; ++p) {
                const int r = (wn * 64 + p * 16 + l16) * LDSPITCH + ko;
                bf[p][0] = bL[r]; bf[p][1] = bL[r + 1];
            }
            #pragma unroll
            for (int i = 0; i < 2; ++i)
                #pragma unroll
                for (int p = 0; p < 4; ++p)
                    acc[i][p] = wmma_f32(af[i], bf[p], acc[i][p]);
        }
        __syncthreads();
    }

    #pragma unroll
    for (int i = 0; i < 2; ++i) {
        #pragma unroll
        for (int p = 0; p < 4; ++p) {
            #pragma unroll
            for (int v = 0; v < 8; ++v) {
                const int grow = row0 + wm * 32 + i * 16 + v + 8 * half;
                out[(size_t)grow * D + col0 + wn * 64 + p * 16 + l16] = acc[i][p][v];
            }
        }
    }
}

// =======================================================================
extern "C" void kernel_launch(void* const* d_in, const int* in_sizes, int n_in,
                              void* d_out, int out_size, void* d_ws, size_t ws_size,
                              hipStream_t stream)
{
    const float* x       = (const float*)d_in[0];
    const float* wq      = (const float*)d_in[1];
    const float* wo      = (const float*)d_in[2];
    const float* k_cache = (const float*)d_in[3];
    const float* v_cache = (const float*)d_in[4];
    const float* cosb    = (const float*)d_in[5];
    const float* sinb    = (const float*)d_in[6];
    // d_in[7] (mask) unused: causal mask computed inline.
    float* out = (float*)d_out;

    char* ws = (char*)d_ws;
    float* q_rope   = (float*)(ws);                                // 32 MB [B][H][S][HD]
    float* k_rope   = (float*)(ws + (size_t)32 * 1024 * 1024);     //  8 MB [B][KV][S][HD]
    float* attn_buf = (float*)(ws + (size_t)40 * 1024 * 1024);     // 32 MB [B][S][D]

    qproj_rope_kernel<<<512, 256, 0, stream>>>(x, wq, cosb, sinb, q_rope);
    krope_kernel<<<4096, 256, 0, stream>>>(k_cache, cosb, sinb, k_rope);
    attn_kernel<<<1024, 256, 0, stream>>>(q_rope, k_rope, v_cache, attn_buf);
    oproj_kernel<<<512, 256, 0, stream>>>(attn_buf, wo, out);
}